// AttentionCapturingEncoderLayer_15307263443056
// MI455X (gfx1250) — compile-verified
//
#include <hip/hip_runtime.h>
#include <hip/hip_bf16.h>
#include <math.h>

// ---------------------------------------------------------------------------
// Transformer encoder layer (pre-norm) for gfx1250 using V_WMMA_F32_16X16X4_F32
// B=8 S=2048 D=384 H=6 HD=64 DFF=1536
// Distance-2 software pipeline with sched_barriers pinning the order
// [wmma-group A ; loads A] [wmma-group B ; loads B] so each WMMA group
// executes while the other set's loads are in flight.
// ---------------------------------------------------------------------------

typedef __attribute__((ext_vector_type(2))) float v2f;
typedef __attribute__((ext_vector_type(8))) float v8f;

#define WMMA_F32(acc, a, b) \
  acc = __builtin_amdgcn_wmma_f32_16x16x4_f32(false, (a), false, (b), (short)0, (acc), false, false)

#define WMMA_GROUP(a, b0, b1, b2, b3)  \
  do {                                 \
    WMMA_F32(acc0, a, b0);             \
    WMMA_F32(acc1, a, b1);             \
    WMMA_F32(acc2, a, b2);             \
    WMMA_F32(acc3, a, b3);             \
  } while (0)

#define SCHED_FENCE() __builtin_amdgcn_sched_barrier(0)

static constexpr int D     = 384;
static constexpr int H     = 6;
static constexpr int HD    = 64;
static constexpr int DFF   = 1536;
static constexpr int S     = 2048;
static constexpr int BATCH = 8;
static constexpr int NTOK  = BATCH * S;   // 16384
static constexpr int QKVD  = 3 * D;       // 1152

static __device__ __forceinline__ v2f ld2(const float* p) {
  return *(const v2f*)p;
}
static __device__ __forceinline__ v2f ld2_nt(const float* p) {
  return __builtin_nontemporal_load((const v2f*)p);
}

// ---------------------------------------------------------------------------
// LayerNorm: one token per 128-thread block (D = 384 = 128*3)
// ---------------------------------------------------------------------------
__global__ __launch_bounds__(128)
void ln_kernel(const float* __restrict__ x, const float* __restrict__ g,
               const float* __restrict__ b, float* __restrict__ y) {
  const int t = blockIdx.x;
  const int tid = threadIdx.x;
  const float* xr = x + (size_t)t * D;
  float v0 = xr[tid], v1 = xr[tid + 128], v2 = xr[tid + 256];

  __shared__ float red[128];
  red[tid] = v0 + v1 + v2;
  __syncthreads();
  for (int o = 64; o > 0; o >>= 1) { if (tid < o) red[tid] += red[tid + o]; __syncthreads(); }
  const float mean = red[0] * (1.0f / 384.0f);
  __syncthreads();
  red[tid] = v0 * v0 + v1 * v1 + v2 * v2;
  __syncthreads();
  for (int o = 64; o > 0; o >>= 1) { if (tid < o) red[tid] += red[tid + o]; __syncthreads(); }
  const float var = red[0] * (1.0f / 384.0f) - mean * mean;
  const float inv = rsqrtf(var + 1e-5f);

  float* yr = y + (size_t)t * D;
  yr[tid]       = (v0 - mean) * inv * g[tid]       + b[tid];
  yr[tid + 128] = (v1 - mean) * inv * g[tid + 128] + b[tid + 128];
  yr[tid + 256] = (v2 - mean) * inv * g[tid + 256] + b[tid + 256];
}

// ---------------------------------------------------------------------------
// Generic GEMM:  C[M,N] = A[M,K] * W[N,K]^T (+bias) (optional ReLU) (+residual)
// Wave computes a 16x64 strip (4 accumulators); block = 4 waves => 64x64 tile.
// grid.x = M/64, grid.y = N/64.  K must be a multiple of 8.
// ---------------------------------------------------------------------------
__global__ __launch_bounds__(128)
void gemm_wmma(const float* __restrict__ A, int lda,
               const float* __restrict__ W, int ldw,
               const float* __restrict__ bias,
               const float* __restrict__ residual,
               float* __restrict__ C, int ldc,
               int K, int relu) {
  const int lane = threadIdx.x & 31;
  const int wave = threadIdx.x >> 5;
  const int m0 = (blockIdx.x * 4 + wave) * 16;
  const int n0 = blockIdx.y * 64;
  const int half = lane >> 4;     // 0: K=k0,k0+1 ; 1: K=k0+2,k0+3
  const int lm = lane & 15;

  const float* aptr = A + (size_t)(m0 + lm) * lda + 2 * half;
  const float* wptr = W + (size_t)(n0 + lm) * ldw + 2 * half;
  const size_t w16 = (size_t)16 * ldw;

  v8f acc0 = {}, acc1 = {}, acc2 = {}, acc3 = {};

  // stage A (k=0) and stage B (k=4)
  v2f aA  = ld2(aptr);
  v2f bA0 = ld2(wptr);
  v2f bA1 = ld2(wptr + w16);
  v2f bA2 = ld2(wptr + 2 * w16);
  v2f bA3 = ld2(wptr + 3 * w16);
  v2f aB  = ld2(aptr + 4);
  v2f bB0 = ld2(wptr + 4);
  v2f bB1 = ld2(wptr + w16 + 4);
  v2f bB2 = ld2(wptr + 2 * w16 + 4);
  v2f bB3 = ld2(wptr + 3 * w16 + 4);

  for (int k0 = 8; k0 < K; k0 += 8) {
    WMMA_GROUP(aA, bA0, bA1, bA2, bA3);
    aA  = ld2(aptr + k0);
    bA0 = ld2(wptr + k0);
    bA1 = ld2(wptr + w16 + k0);
    bA2 = ld2(wptr + 2 * w16 + k0);
    bA3 = ld2(wptr + 3 * w16 + k0);
    SCHED_FENCE();
    WMMA_GROUP(aB, bB0, bB1, bB2, bB3);
    aB  = ld2(aptr + k0 + 4);
    bB0 = ld2(wptr + k0 + 4);
    bB1 = ld2(wptr + w16 + k0 + 4);
    bB2 = ld2(wptr + 2 * w16 + k0 + 4);
    bB3 = ld2(wptr + 3 * w16 + k0 + 4);
    SCHED_FENCE();
  }
  WMMA_GROUP(aA, bA0, bA1, bA2, bA3);
  WMMA_GROUP(aB, bB0, bB1, bB2, bB3);

  v8f accs[4] = {acc0, acc1, acc2, acc3};
#pragma unroll
  for (int j = 0; j < 4; ++j) {
    const int n = n0 + j * 16 + lm;
    const float bv = bias ? bias[n] : 0.0f;
#pragma unroll
    for (int r = 0; r < 8; ++r) {
      const int m = m0 + r + 8 * half;
      float v = accs[j][r] + bv;
      if (relu) v = fmaxf(v, 0.0f);
      const size_t idx = (size_t)m * ldc + n;
      if (residual) v += residual[idx];
      C[idx] = v;
    }
  }
}

// ---------------------------------------------------------------------------
// Scores: per (b,h)  P[S,S] = (Q * K^T) / 8    (raw, softmax applied later)
// grid = (S/64, S/64, B*H).  Non-temporal stores (805 MB stream > L2).
// ---------------------------------------------------------------------------
__global__ __launch_bounds__(128)
void scores_wmma(const float* __restrict__ qkv, float* __restrict__ attn) {
  const int z = blockIdx.z;
  const int b = z / H, h = z % H;
  const int lane = threadIdx.x & 31;
  const int wave = threadIdx.x >> 5;
  const int m0 = (blockIdx.x * 4 + wave) * 16;
  const int n0 = blockIdx.y * 64;
  const int half = lane >> 4;
  const int lm = lane & 15;

  const float* qbase = qkv + (size_t)(b * S) * QKVD + h * HD;
  const float* kbase = qbase + D;
  const float* qptr = qbase + (size_t)(m0 + lm) * QKVD + 2 * half;
  const float* kptr = kbase + (size_t)(n0 + lm) * QKVD + 2 * half;
  const size_t k16 = (size_t)16 * QKVD;

  v8f acc0 = {}, acc1 = {}, acc2 = {}, acc3 = {};

  v2f aA  = ld2(qptr);
  v2f bA0 = ld2(kptr);
  v2f bA1 = ld2(kptr + k16);
  v2f bA2 = ld2(kptr + 2 * k16);
  v2f bA3 = ld2(kptr + 3 * k16);
  v2f aB  = ld2(qptr + 4);
  v2f bB0 = ld2(kptr + 4);
  v2f bB1 = ld2(kptr + k16 + 4);
  v2f bB2 = ld2(kptr + 2 * k16 + 4);
  v2f bB3 = ld2(kptr + 3 * k16 + 4);

#pragma unroll
  for (int k0 = 8; k0 < HD; k0 += 8) {
    WMMA_GROUP(aA, bA0, bA1, bA2, bA3);
    aA  = ld2(qptr + k0);
    bA0 = ld2(kptr + k0);
    bA1 = ld2(kptr + k16 + k0);
    bA2 = ld2(kptr + 2 * k16 + k0);
    bA3 = ld2(kptr + 3 * k16 + k0);
    SCHED_FENCE();
    WMMA_GROUP(aB, bB0, bB1, bB2, bB3);
    aB  = ld2(qptr + k0 + 4);
    bB0 = ld2(kptr + k0 + 4);
    bB1 = ld2(kptr + k16 + k0 + 4);
    bB2 = ld2(kptr + 2 * k16 + k0 + 4);
    bB3 = ld2(kptr + 3 * k16 + k0 + 4);
    SCHED_FENCE();
  }
  WMMA_GROUP(aA, bA0, bA1, bA2, bA3);
  WMMA_GROUP(aB, bB0, bB1, bB2, bB3);

  float* pbase = attn + (size_t)z * S * S;
  v8f accs[4] = {acc0, acc1, acc2, acc3};
#pragma unroll
  for (int j = 0; j < 4; ++j) {
    const int n = n0 + j * 16 + lm;
#pragma unroll
    for (int r = 0; r < 8; ++r) {
      const int m = m0 + r + 8 * half;
      __builtin_nontemporal_store(accs[j][r] * 0.125f, &pbase[(size_t)m * S + n]);
    }
  }
}

// ---------------------------------------------------------------------------
// Row softmax in place over attn (B*H*S rows of length S=2048). 256 thr/row.
// ---------------------------------------------------------------------------
__global__ __launch_bounds__(256)
void softmax_rows(float* __restrict__ attn) {
  float* p = attn + (size_t)blockIdx.x * S;
  const int tid = threadIdx.x;
  float v[8];
  float mx = -INFINITY;
#pragma unroll
  for (int i = 0; i < 8; ++i) {
    v[i] = __builtin_nontemporal_load(&p[tid + i * 256]);
    mx = fmaxf(mx, v[i]);
  }

  __shared__ float red[256];
  red[tid] = mx;
  __syncthreads();
  for (int o = 128; o > 0; o >>= 1) { if (tid < o) red[tid] = fmaxf(red[tid], red[tid + o]); __syncthreads(); }
  mx = red[0];
  __syncthreads();

  float s = 0.0f;
#pragma unroll
  for (int i = 0; i < 8; ++i) { v[i] = __expf(v[i] - mx); s += v[i]; }
  red[tid] = s;
  __syncthreads();
  for (int o = 128; o > 0; o >>= 1) { if (tid < o) red[tid] += red[tid + o]; __syncthreads(); }
  const float inv = 1.0f / red[0];
#pragma unroll
  for (int i = 0; i < 8; ++i)
    __builtin_nontemporal_store(v[i] * inv, &p[tid + i * 256]);
}

// ---------------------------------------------------------------------------
// Context: per (b,h)  ctx[S,HD] = P[S,S] * V[S,HD], written to [B,S,D] layout
// grid = (S/64, 1, B*H); each wave covers the full HD=64 width.
// P rows are read exactly once -> non-temporal loads.  V stays L2-resident.
// ---------------------------------------------------------------------------
__global__ __launch_bounds__(128)
void ctx_wmma(const float* __restrict__ attn, const float* __restrict__ qkv,
              float* __restrict__ ctx) {
  const int z = blockIdx.z;
  const int b = z / H, h = z % H;
  const int lane = threadIdx.x & 31;
  const int wave = threadIdx.x >> 5;
  const int m0 = (blockIdx.x * 4 + wave) * 16;
  const int half = lane >> 4;
  const int lm = lane & 15;

  const float* pptr  = attn + (size_t)z * S * S + (size_t)(m0 + lm) * S + 2 * half;
  const float* vbase = qkv + (size_t)(b * S) * QKVD + 2 * D + h * HD + (size_t)(2 * half) * QKVD;

  v8f acc0 = {}, acc1 = {}, acc2 = {}, acc3 = {};

  // stage A (k=0)
  v2f aA = ld2_nt(pptr);
  v2f bA0 = {vbase[lm],      vbase[QKVD + lm]};
  v2f bA1 = {vbase[lm + 16], vbase[QKVD + lm + 16]};
  v2f bA2 = {vbase[lm + 32], vbase[QKVD + lm + 32]};
  v2f bA3 = {vbase[lm + 48], vbase[QKVD + lm + 48]};
  // stage B (k=4)
  const float* v4 = vbase + (size_t)4 * QKVD;
  v2f aB = ld2_nt(pptr + 4);
  v2f bB0 = {v4[lm],      v4[QKVD + lm]};
  v2f bB1 = {v4[lm + 16], v4[QKVD + lm + 16]};
  v2f bB2 = {v4[lm + 32], v4[QKVD + lm + 32]};
  v2f bB3 = {v4[lm + 48], v4[QKVD + lm + 48]};

  for (int k0 = 8; k0 < S; k0 += 8) {
    WMMA_GROUP(aA, bA0, bA1, bA2, bA3);
    {
      const float* w0 = vbase + (size_t)k0 * QKVD;
      aA  = ld2_nt(pptr + k0);
      bA0 = {w0[lm],      w0[QKVD + lm]};
      bA1 = {w0[lm + 16], w0[QKVD + lm + 16]};
      bA2 = {w0[lm + 32], w0[QKVD + lm + 32]};
      bA3 = {w0[lm + 48], w0[QKVD + lm + 48]};
    }
    SCHED_FENCE();
    WMMA_GROUP(aB, bB0, bB1, bB2, bB3);
    {
      const float* w0 = vbase + (size_t)(k0 + 4) * QKVD;
      aB  = ld2_nt(pptr + k0 + 4);
      bB0 = {w0[lm],      w0[QKVD + lm]};
      bB1 = {w0[lm + 16], w0[QKVD + lm + 16]};
      bB2 = {w0[lm + 32], w0[QKVD + lm + 32]};
      bB3 = {w0[lm + 48], w0[QKVD + lm + 48]};
    }
    SCHED_FENCE();
  }
  WMMA_GROUP(aA, bA0, bA1, bA2, bA3);
  WMMA_GROUP(aB, bB0, bB1, bB2, bB3);

  float* cbase = ctx + (size_t)(b * S + m0) * D + h * HD;
  v8f accs[4] = {acc0, acc1, acc2, acc3};
#pragma unroll
  for (int j = 0; j < 4; ++j) {
    const int n = j * 16 + lm;
#pragma unroll
    for (int r = 0; r < 8; ++r) {
      const int m = r + 8 * half;
      cbase[(size_t)m * D + n] = accs[j][r];
    }
  }
}

// ---------------------------------------------------------------------------
// Host-side launch
// ---------------------------------------------------------------------------
extern "C" void kernel_launch(void* const* d_in, const int* in_sizes, int n_in,
                              void* d_out, int out_size, void* d_ws, size_t ws_size,
                              hipStream_t stream) {
  const float* src        = (const float*)d_in[0];
  const float* in_proj_w  = (const float*)d_in[1];
  const float* in_proj_b  = (const float*)d_in[2];
  const float* out_proj_w = (const float*)d_in[3];
  const float* out_proj_b = (const float*)d_in[4];
  const float* w1         = (const float*)d_in[5];
  const float* b1         = (const float*)d_in[6];
  const float* w2         = (const float*)d_in[7];
  const float* b2         = (const float*)d_in[8];
  const float* ln1_g      = (const float*)d_in[9];
  const float* ln1_b      = (const float*)d_in[10];
  const float* ln2_g      = (const float*)d_in[11];
  const float* ln2_b      = (const float*)d_in[12];

  float* out_x = (float*)d_out;                    // [8,2048,384]
  float* attn  = out_x + (size_t)NTOK * D;         // [8,6,2048,2048]

  float* ws = (float*)d_ws;
  // Aliased layout (peak ~176 MB of float scratch):
  float* xin  = ws;                                // [16384,384] (LN1 out; freed after QKV)
  float* ctx  = ws;                                // reuses xin region after QKV GEMM
  float* qkvb = ws + (size_t)NTOK * D;             // [16384,1152] (freed after ctx)
  float* x1   = ws + (size_t)NTOK * D;             // reuses qkv region
  float* xff  = x1 + (size_t)NTOK * D;             // [16384,384]
  float* ff   = xff + (size_t)NTOK * D;            // [16384,1536]

  // 1) LN1
  ln_kernel<<<NTOK, 128, 0, stream>>>(src, ln1_g, ln1_b, xin);

  // 2) QKV projection: [16384,1152] = xin * in_proj_w^T + b
  gemm_wmma<<<dim3(NTOK / 64, QKVD / 64), 128, 0, stream>>>(
      xin, D, in_proj_w, D, in_proj_b, nullptr, qkvb, QKVD, D, 0);

  // 3) Raw attention scores (scaled), staged into the attn output region
  scores_wmma<<<dim3(S / 64, S / 64, BATCH * H), 128, 0, stream>>>(qkvb, attn);

  // 4) Softmax in place -> final attn_weights output
  softmax_rows<<<BATCH * H * S, 256, 0, stream>>>(attn);

  // 5) Context: P * V -> ctx [B,S,D]
  ctx_wmma<<<dim3(S / 64, 1, BATCH * H), 128, 0, stream>>>(attn, qkvb, ctx);

  // 6) Out projection + residual: x1 = src + ctx * Wo^T + bo
  gemm_wmma<<<dim3(NTOK / 64, D / 64), 128, 0, stream>>>(
      ctx, D, out_proj_w, D, out_proj_b, src, x1, D, D, 0);

  // 7) LN2
  ln_kernel<<<NTOK, 128, 0, stream>>>(x1, ln2_g, ln2_b, xff);

  // 8) FFN1 with ReLU: ff = relu(xff * w1^T + b1)
  gemm_wmma<<<dim3(NTOK / 64, DFF / 64), 128, 0, stream>>>(
      xff, D, w1, D, b1, nullptr, ff, DFF, D, 1);

  // 9) FFN2 + residual -> final output x
  gemm_wmma<<<dim3(NTOK / 64, D / 64), 128, 0, stream>>>(
      ff, DFF, w2, DFF, b2, x1, out_x, D, DFF, 0);
}